// StructureLearner_34531537060042
// MI455X (gfx1250) — compile-verified
//
#include <hip/hip_runtime.h>
#include <stdint.h>

typedef float v4f __attribute__((ext_vector_type(4)));

// Problem constants (from the reference)
#define BB        128
#define DDIM      512
#define DD        (DDIM * DDIM)      // 262144 elements per matrix
#define KSEL      26214              // max(1, int(0.1 * 512 * 512))
#define TPB       512                // 16 wave32s per block
#define CHUNK     4096               // floats staged per matrix per chunk (16 KB)
#define NCHUNK    (DD / CHUNK)       // 64 chunks
#define VEC_SLOTS (CHUNK / 4)        // 1024 float4 slots per matrix chunk

// Map float -> uint32 key with the same total order (larger float => larger key)
__device__ __forceinline__ uint32_t f2k(float x) {
    uint32_t u = __float_as_uint(x);
    return (u & 0x80000000u) ? ~u : (u | 0x80000000u);
}

__device__ __forceinline__ float sigmoidf_(float x) {
    return 1.0f / (1.0f + __expf(-x));
}

// Low 32 bits of a generic LDS pointer == LDS byte address (ISA 10.2)
__device__ __forceinline__ uint32_t lds_lo32(const void* p) {
    return (uint32_t)(uintptr_t)p;
}

// CDNA5 async DMA: global -> LDS, 16 bytes per lane, ASYNCcnt-tracked
__device__ __forceinline__ void async_copy_b128(uint32_t lds_byte_addr, const float* gaddr) {
    asm volatile("global_load_async_to_lds_b128 %0, %1, off"
                 :: "v"(lds_byte_addr), "v"(gaddr)
                 : "memory");
}
__device__ __forceinline__ void wait_async_le0() {
    asm volatile("s_wait_asynccnt 0" ::: "memory");
}
// Each wave issues exactly 4 async instructions per chunk
// (VEC_SLOTS/TPB = 2 iterations x 2 matrices), so <=4 means "previous chunk done".
__device__ __forceinline__ void wait_async_le4() {
    asm volatile("s_wait_asynccnt 4" ::: "memory");
}

__global__ __launch_bounds__(TPB)
void structure_learner_kernel(const int* __restrict__ env_idx,
                              const float* __restrict__ A_base,
                              const float* __restrict__ A_deltas,
                              float* __restrict__ outA,
                              float* __restrict__ outLogits,
                              float* __restrict__ outSoft) {
    // Ping-pong staging buffers: [buf][matrix][CHUNK]
    __shared__ __align__(16) float stage[2][2][CHUNK];     // 64 KB
    __shared__ uint32_t hist[256];
    __shared__ uint32_t sh_prefix;      // radix-select prefix -> exact k-th key
    __shared__ uint32_t sh_remaining;   // rank remaining inside current prefix
    __shared__ uint32_t sh_tiecnt;      // tie-inclusion counter (final pass)

    const int b   = blockIdx.x;         // one workgroup per sample
    const int tid = threadIdx.x;
    const int env = env_idx[b];

    const float* base  = A_base;
    const float* delta = A_deltas + (size_t)env * DD;
    v4f* oA4 = (v4f*)(outA      + (size_t)b * DD);
    v4f* oL4 = (v4f*)(outLogits + (size_t)b * DD);
    v4f* oS4 = (v4f*)(outSoft   + (size_t)b * DD);

    if (tid == 0) { sh_prefix = 0u; sh_remaining = KSEL; sh_tiecnt = 0u; }

    uint32_t sb_addr[2], sd_addr[2];
    sb_addr[0] = lds_lo32(stage[0][0]);  sd_addr[0] = lds_lo32(stage[0][1]);
    sb_addr[1] = lds_lo32(stage[1][0]);  sd_addr[1] = lds_lo32(stage[1][1]);

    // Issue one chunk's DMA (4 async instructions per wave)
    auto stage_chunk = [&](int buf, int c) {
        const int off = c * CHUNK;
        for (int t = tid; t < VEC_SLOTS; t += TPB) {
            async_copy_b128(sb_addr[buf] + t * 16, base  + off + t * 4);
            async_copy_b128(sd_addr[buf] + t * 16, delta + off + t * 4);
        }
    };

    // ---- 4 radix-select passes (MSB digit first). Pass 0 fuses the
    // ---- A_logits / A_soft streaming stores (the HBM-bound work).
    for (int pass = 0; pass < 4; ++pass) {
        const int      shift  = 24 - 8 * pass;
        const uint32_t maskHi = (pass == 0) ? 0u : (0xFFFFFFFFu << (shift + 8));

        for (int i = tid; i < 256; i += TPB) hist[i] = 0u;
        __syncthreads();
        const uint32_t prefix = sh_prefix;

        stage_chunk(0, 0);
        for (int c = 0; c < NCHUNK; ++c) {
            const int cur = c & 1;
            if (c + 1 < NCHUNK) { stage_chunk(cur ^ 1, c + 1); wait_async_le4(); }
            else                { wait_async_le0(); }
            __syncthreads();                       // all waves' DMA for chunk c visible

            const v4f* pb = (const v4f*)stage[cur][0];
            const v4f* pd = (const v4f*)stage[cur][1];
            for (int t = tid; t < VEC_SLOTS; t += TPB) {
                const v4f lg = pb[t] + pd[t];      // ds_load_b128 x2
                if (pass == 0) {
                    v4f sg;
                    #pragma unroll
                    for (int j = 0; j < 4; ++j) sg[j] = sigmoidf_(lg[j]);
                    __builtin_nontemporal_store(lg, oL4 + (size_t)c * VEC_SLOTS + t);
                    __builtin_nontemporal_store(sg, oS4 + (size_t)c * VEC_SLOTS + t);
                }
                #pragma unroll
                for (int j = 0; j < 4; ++j) {
                    const uint32_t key = f2k(lg[j]);
                    if ((key & maskHi) == (prefix & maskHi))
                        atomicAdd(&hist[(key >> shift) & 0xFFu], 1u);
                }
            }
            __syncthreads();                       // chunk consumed before buffer reuse
        }

        // Pick the digit containing the k-th largest (scan bins high -> low)
        if (tid == 0) {
            uint32_t rem = sh_remaining;
            uint32_t cum = 0u;
            int d = 255;
            for (; d > 0; --d) {
                const uint32_t cc = hist[d];
                if (cum + cc >= rem) break;
                cum += cc;
            }
            sh_prefix    = prefix | ((uint32_t)d << shift);
            sh_remaining = rem - cum;
        }
        __syncthreads();
    }

    const uint32_t kth  = sh_prefix;     // exact k-th largest key
    const uint32_t need = sh_remaining;  // how many elements == kth to keep

    // ---- Final pass: stream A (sigmoid at selected positions, 0 elsewhere)
    stage_chunk(0, 0);
    for (int c = 0; c < NCHUNK; ++c) {
        const int cur = c & 1;
        if (c + 1 < NCHUNK) { stage_chunk(cur ^ 1, c + 1); wait_async_le4(); }
        else                { wait_async_le0(); }
        __syncthreads();

        const v4f* pb = (const v4f*)stage[cur][0];
        const v4f* pd = (const v4f*)stage[cur][1];
        for (int t = tid; t < VEC_SLOTS; t += TPB) {
            const v4f lg = pb[t] + pd[t];
            v4f out;
            #pragma unroll
            for (int j = 0; j < 4; ++j) {
                const uint32_t key = f2k(lg[j]);
                float v = 0.0f;
                if (key > kth) {
                    v = sigmoidf_(lg[j]);
                } else if (key == kth) {
                    if (atomicAdd(&sh_tiecnt, 1u) < need) v = sigmoidf_(lg[j]);
                }
                out[j] = v;
            }
            __builtin_nontemporal_store(out, oA4 + (size_t)c * VEC_SLOTS + t);
        }
        __syncthreads();
    }
}

extern "C" void kernel_launch(void* const* d_in, const int* in_sizes, int n_in,
                              void* d_out, int out_size, void* d_ws, size_t ws_size,
                              hipStream_t stream) {
    // Inputs (setup_inputs order): z_s (unused), env_idx, A_base, A_deltas
    const int*   env_idx  = (const int*)d_in[1];
    const float* A_base   = (const float*)d_in[2];
    const float* A_deltas = (const float*)d_in[3];

    // Outputs concatenated flat in return order: A, A_logits, A_soft
    float* out  = (float*)d_out;
    float* outA = out;
    float* outL = out + (size_t)BB * DD;
    float* outS = out + 2 * (size_t)BB * DD;

    structure_learner_kernel<<<dim3(BB), dim3(TPB), 0, stream>>>(
        env_idx, A_base, A_deltas, outA, outL, outS);
}